// GoLTransBlock_89163521065545
// MI455X (gfx1250) — compile-verified
//
#include <hip/hip_runtime.h>
#include <math.h>

#define BB 8
#define CC 256
#define HH 128
#define WW 128
#define GG 64
#define MM 4
#define KK 16
#define KNN8 8

typedef __attribute__((ext_vector_type(2))) float v2f;
typedef __attribute__((ext_vector_type(8))) float v8f;

__device__ __forceinline__ float waveRedSum(float s) {
#pragma unroll
  for (int off = 16; off > 0; off >>= 1) s += __shfl_xor(s, off, 32);
  return s;
}

// CDNA5 async global->LDS copy: one wave instruction moves 32 lanes x 16B.
// vdst = VGPR with LDS byte address (low 32 bits of flat shared pointer),
// vaddr = 64-bit global address VGPR pair, no SADDR ("off"). Tracked by ASYNCcnt.
__device__ __forceinline__ void asyncRow128(const float* g, float* l) {
  asm volatile("global_load_async_to_lds_b128 %0, %1, off"
               :: "v"((unsigned)(size_t)l), "v"((unsigned long long)g)
               : "memory");
}
__device__ __forceinline__ void asyncWaitAll() {
  asm volatile("s_wait_asynccnt 0x0" ::: "memory");
}

// block reductions for exactly 128 threads
__device__ __forceinline__ float bSum128(float v, float* red, int tid) {
  __syncthreads();
  red[tid] = v; __syncthreads();
#pragma unroll
  for (int s = 64; s > 0; s >>= 1) { if (tid < s) red[tid] += red[tid + s]; __syncthreads(); }
  float r = red[0]; __syncthreads();
  return r;
}
__device__ __forceinline__ float bMax128(float v, float* red, int tid) {
  __syncthreads();
  red[tid] = v; __syncthreads();
#pragma unroll
  for (int s = 64; s > 0; s >>= 1) { if (tid < s) red[tid] = fmaxf(red[tid], red[tid + s]); __syncthreads(); }
  float r = red[0]; __syncthreads();
  return r;
}

// ---------------- Kernel 1: X_node(b,c,h) = mean_w x ----------------
__global__ void k1_rowmean(const float* __restrict__ x, float* __restrict__ Xnode) {
  const int bc = blockIdx.x;                       // b*C + c
  const int wv = threadIdx.x >> 5, lane = threadIdx.x & 31;
  const float4* xp = (const float4*)(x + (size_t)bc * (HH * WW));
  for (int r = 0; r < 16; ++r) {
    const int h = wv * 16 + r;
    float4 v = xp[h * (WW / 4) + lane];
    float s = v.x + v.y + v.z + v.w;
    s = waveRedSum(s);
    if (lane == 0) Xnode[(size_t)bc * HH + h] = s * (1.0f / 128.0f);
  }
}

// ---------------- Kernel 2: GCN + all small tensors (1 block per b, 128 thr) ----------------
__global__ void k2_small(const float* __restrict__ U, const float* __restrict__ S,
                         const float* __restrict__ V, const float* __restrict__ w_row,
                         const float* __restrict__ w_col, const float* __restrict__ boast,
                         const float* __restrict__ reduce_w, const float* __restrict__ gcn_w,
                         const float* __restrict__ gcn_b, const float* __restrict__ expand_w,
                         const float* __restrict__ Xnode,
                         float* __restrict__ att_row, float* __restrict__ att_col,
                         float* __restrict__ topo,
                         float* __restrict__ a_row, float* __restrict__ a_col,
                         float* __restrict__ a_chan, float* __restrict__ sig_ch,
                         float* __restrict__ sm_c, float* __restrict__ sig_h) {
  extern __shared__ float sm2[];
  float* Un     = sm2;              // 2048
  float* gcnw   = Un + 2048;        // 4096
  float* sup    = gcnw + 4096;      // 8192
  float* outg   = sup + 8192;       // 8192
  float* dinv   = outg + 8192;      // 128
  float* usum   = dinv + 128;       // 128
  float* tmn    = usum + 128;       // 128
  float* absexp = tmn + 128;        // 64
  float* sumexp = absexp + 64;      // 64
  float* cmg    = sumexp + 64;      // 64
  float* zt     = cmg + 64;         // 256
  float* zx     = zt + 256;         // 256
  float* zk     = zx + 256;         // 16
  float* red    = zk + 16;          // 128
  const int b = blockIdx.x;
  const int tid = threadIdx.x;      // doubles as h (and as w)

  // Phase A: normalize U rows, usum, stage gcn_w, expand_w row stats
  float ur[KK]; float nrm = 0.f, us = 0.f;
#pragma unroll
  for (int k = 0; k < KK; ++k) {
    ur[k] = U[((size_t)b * HH + tid) * KK + k];
    nrm += ur[k] * ur[k]; us += ur[k];
  }
  nrm = fmaxf(sqrtf(nrm), 1e-12f);
#pragma unroll
  for (int k = 0; k < KK; ++k) { ur[k] /= nrm; Un[tid * KK + k] = ur[k]; }
  usum[tid] = us;
  for (int idx = tid; idx < GG * GG; idx += 128) gcnw[idx] = gcn_w[idx];
  if (tid < GG) {
    float a = 0.f, s2 = 0.f;
#pragma unroll
    for (int mi = 0; mi < MM; ++mi) { float v = expand_w[tid * MM + mi]; a += fabsf(v); s2 += v; }
    absexp[tid] = a; sumexp[tid] = s2;
  }
  __syncthreads();

  // Phase B: similarity row + top-8 (jax tie-break: earlier index first on equal)
  float tv[KNN8]; int ti[KNN8];
#pragma unroll
  for (int t = 0; t < KNN8; ++t) { tv[t] = -3.0e38f; ti[t] = 0; }
  for (int j = 0; j < HH; ++j) {
    float s = 0.f;
#pragma unroll
    for (int k = 0; k < KK; ++k) s += ur[k] * Un[j * KK + k];
    if (s > tv[KNN8 - 1]) {
      int p = KNN8 - 1;
      while (p > 0 && tv[p - 1] < s) { tv[p] = tv[p - 1]; ti[p] = ti[p - 1]; --p; }
      tv[p] = s; ti[p] = j;
    }
  }
  float deg = 0.f;
#pragma unroll
  for (int t = 0; t < KNN8; ++t) deg += tv[t];
  float dh = 1.0f / sqrtf(fmaxf(deg, 1e-30f));
  dinv[tid] = dh;

  // X_red row (registers) -> support row (LDS)
  float xr[GG];
#pragma unroll
  for (int gi = 0; gi < GG; ++gi) {
    float a = 0.f;
#pragma unroll
    for (int mi = 0; mi < MM; ++mi)
      a += Xnode[((size_t)b * CC + gi * MM + mi) * HH + tid] * reduce_w[gi * MM + mi];
    xr[gi] = a;
  }
  for (int d = 0; d < GG; ++d) {
    float a = 0.f;
#pragma unroll
    for (int gi = 0; gi < GG; ++gi) a += xr[gi] * gcnw[gi * GG + d];
    sup[tid * GG + d] = a;
  }
  __syncthreads();

  // Phase C: out = relu(adj @ support + b) via 8-neighbor gather
  float coef[KNN8];
#pragma unroll
  for (int t = 0; t < KNN8; ++t) coef[t] = tv[t] * dh * dinv[ti[t]];
  for (int d = 0; d < GG; ++d) {
    float a = gcn_b[d];
#pragma unroll
    for (int t = 0; t < KNN8; ++t) a += coef[t] * sup[ti[t] * GG + d];
    outg[tid * GG + d] = fmaxf(a, 0.f);
  }
  __syncthreads();

  // Phase D: topo(b,c,h) writeout
  for (int idx = tid; idx < CC * HH; idx += 128) {
    int c = idx >> 7, hh2 = idx & 127;
    topo[((size_t)b * CC + c) * HH + hh2] = outg[hh2 * GG + (c >> 2)] * expand_w[c];
  }
  // topo_map (normalized)
  float tmv = 0.f;
#pragma unroll
  for (int gi = 0; gi < GG; ++gi) tmv += fabsf(outg[tid * GG + gi]) * absexp[gi];
  float tmax = bMax128(tmv, red, tid);
  tmn[tid] = tmv / fmaxf(tmax, 1e-6f);
  // sig_h(b,h) = sigmoid(mean_c topo)
  float shv = 0.f;
#pragma unroll
  for (int gi = 0; gi < GG; ++gi) shv += outg[tid * GG + gi] * sumexp[gi];
  shv *= (1.0f / (float)CC);
  sig_h[b * HH + tid] = 1.0f / (1.0f + expf(-shv));
  // column means of outg
  if (tid < GG) {
    float a = 0.f;
    for (int hh2 = 0; hh2 < HH; ++hh2) a += outg[hh2 * GG + tid];
    cmg[tid] = a * (1.0f / (float)HH);
  }
  __syncthreads();
  // z_topo (zt) and z_x (zx)
  for (int c = tid; c < CC; c += 128) {
    zt[c] = cmg[c >> 2] * expand_w[c];
    float a = 0.f;
    for (int hh2 = 0; hh2 < HH; ++hh2) a += Xnode[((size_t)b * CC + c) * HH + hh2];
    zx[c] = a * (1.0f / (float)HH);
  }
  __syncthreads();
  // sm_c = softmax_c(z_topo)
  {
    float lm = fmaxf(zt[tid], zt[tid + 128]);
    float zmax = bMax128(lm, red, tid);
    float e0 = expf(zt[tid] - zmax), e1 = expf(zt[tid + 128] - zmax);
    float zsum = bSum128(e0 + e1, red, tid);
    sm_c[b * CC + tid] = e0 / zsum;
    sm_c[b * CC + tid + 128] = e1 / zsum;
  }
  __syncthreads();
  if (tid < KK) {
    float a = 0.f;
    for (int c = 0; c < CC; ++c)
      a += (zx[c] + 0.5f * zt[c]) * V[((size_t)b * CC + c) * KK + tid];
    zk[tid] = a;
  }
  __syncthreads();
  float Ssum = 0.f, zks = 0.f;
#pragma unroll
  for (int j = 0; j < KK; ++j) { Ssum += S[b * KK + j]; zks += zk[j]; }
  // sig_ch(b,c) = sigmoid((V.zk * Ssum + V.S * zksum)/32)
  for (int c = tid; c < CC; c += 128) {
    float vz = 0.f, vs = 0.f;
#pragma unroll
    for (int kk2 = 0; kk2 < KK; ++kk2) {
      float vv = V[((size_t)b * CC + c) * KK + kk2];
      vz += vv * zk[kk2]; vs += vv * S[b * KK + kk2];
    }
    float cr = (vz * Ssum + vs * zks) * (1.0f / 32.0f);
    sig_ch[b * CC + c] = 1.0f / (1.0f + expf(-cr));
  }
  // att_row (thread = row h); const topo term cancels in softmax over w
  {
    float uh = usum[tid];
    float mx = -3.0e38f;
    for (int w2 = 0; w2 < WW; ++w2) mx = fmaxf(mx, uh * w_row[w2]);
    float ssum = 0.f;
    for (int w2 = 0; w2 < WW; ++w2) ssum += expf(uh * w_row[w2] - mx);
    float inv = 1.0f / ssum;
    for (int w2 = 0; w2 < WW; ++w2)
      att_row[((size_t)b * HH + tid) * WW + w2] = expf(uh * w_row[w2] - mx) * inv;
  }
  // att_col (thread = col w), softmax over h
  {
    float wc = w_col[tid];
    float mx = -3.0e38f;
    for (int hh2 = 0; hh2 < HH; ++hh2) mx = fmaxf(mx, usum[hh2] * wc + 0.5f * tmn[hh2]);
    float ssum = 0.f;
    for (int hh2 = 0; hh2 < HH; ++hh2) ssum += expf(usum[hh2] * wc + 0.5f * tmn[hh2] - mx);
    float inv = 1.0f / ssum;
    for (int hh2 = 0; hh2 < HH; ++hh2)
      att_col[((size_t)b * HH + hh2) * WW + tid] = expf(usum[hh2] * wc + 0.5f * tmn[hh2] - mx) * inv;
  }
  // w4 = softmax(t(b)*boast) over 512, split into a_row/a_col/a_chan
  {
    float twr = bSum128(w_row[tid], red, tid);
    float twc = bSum128(w_col[tid], red, tid);
    float tb_ = Ssum + twr + twc;
    float l0 = tb_ * boast[tid],       l1 = tb_ * boast[tid + 128];
    float l2 = tb_ * boast[tid + 256], l3 = tb_ * boast[tid + 384];
    float lm = fmaxf(fmaxf(l0, l1), fmaxf(l2, l3));
    float bmax = bMax128(lm, red, tid);
    float e0 = expf(l0 - bmax), e1 = expf(l1 - bmax), e2 = expf(l2 - bmax), e3 = expf(l3 - bmax);
    float bsum = bSum128(e0 + e1 + e2 + e3, red, tid);
    float inv = 1.0f / bsum;
    a_row[b * HH + tid] = e0 * inv;
    a_col[b * WW + tid] = e1 * inv;
    a_chan[b * CC + tid] = e2 * inv;
    a_chan[b * CC + 128 + tid] = e3 * inv;
  }
}

// ---------------- Kernel 3: R and Cctx (fused vectorized pass over x) ----------------
__global__ void k3_attred(const float* __restrict__ x, const float* __restrict__ att_row,
                          const float* __restrict__ att_col,
                          float* __restrict__ Rm, float* __restrict__ Cm) {
  __shared__ float colred[8 * 128];
  const int bc = blockIdx.x;
  const int b = bc >> 8;
  const int wv = threadIdx.x >> 5, lane = threadIdx.x & 31;
  const float4* xp  = (const float4*)(x + (size_t)bc * (HH * WW));
  const float4* arp = (const float4*)(att_row + (size_t)b * (HH * WW));
  const float4* acp = (const float4*)(att_col + (size_t)b * (HH * WW));
  float ca0 = 0.f, ca1 = 0.f, ca2 = 0.f, ca3 = 0.f;
  for (int r = 0; r < 16; ++r) {
    const int h = wv * 16 + r;
    const int o = h * (WW / 4) + lane;
    float4 xv = xp[o];
    float4 ar = arp[o];
    float4 ac = acp[o];
    float s = xv.x * ar.x + xv.y * ar.y + xv.z * ar.z + xv.w * ar.w;
    s = waveRedSum(s);
    if (lane == 0) Rm[(size_t)bc * HH + h] = s;
    ca0 += xv.x * ac.x; ca1 += xv.y * ac.y;
    ca2 += xv.z * ac.z; ca3 += xv.w * ac.w;
  }
  colred[wv * 128 + lane * 4 + 0] = ca0;
  colred[wv * 128 + lane * 4 + 1] = ca1;
  colred[wv * 128 + lane * 4 + 2] = ca2;
  colred[wv * 128 + lane * 4 + 3] = ca3;
  __syncthreads();
  const int t = threadIdx.x;
  if (t < 128) {
    float s = 0.f;
#pragma unroll
    for (int v = 0; v < 8; ++v) s += colred[v * 128 + t];
    Cm[(size_t)bc * WW + t] = s;
  }
}

// ---------------- Kernel 4: dual f32 WMMA GEMM + fused epilogue ----------------
// LDS pitch 132: rows stay 16B-aligned (528B) for ASYNC_TO_LDS_B128 and the
// fragment fetch pattern (16 rows x {k,k+1}/{k+2,k+3}) hits all 64 banks once.
__global__ void k4_gemm_out(const float* __restrict__ x, const float* __restrict__ Gm,
                            const float* __restrict__ Rm, const float* __restrict__ Cm,
                            const float* __restrict__ topo,
                            const float* __restrict__ a_row, const float* __restrict__ a_col,
                            const float* __restrict__ a_chan, const float* __restrict__ sig_ch,
                            const float* __restrict__ sm_c, const float* __restrict__ sig_h,
                            const float* __restrict__ res_scale, float* __restrict__ out) {
  extern __shared__ float sm4[];
  const int PITCH = 132;
  float* Gs = sm4;                       // 128*132
  float* Ar = Gs + 128 * PITCH;          // 16*132
  float* Ac = Ar + 16 * PITCH;           // 16*132
  float* achsig = Ac + 16 * PITCH;       // 16
  float* tbv = achsig + 16;              // 16
  const int hb = blockIdx.x;             // output row (128 pixels == one h row)
  const int ct = blockIdx.y;             // channel tile
  const int b  = blockIdx.z;
  const int c0 = ct * 16;
  const int tid = threadIdx.x;
  const int lane = tid & 31, wv = tid >> 5;

  // ---- async global->LDS staging (ASYNCcnt), no VGPR round-trip ----
  const float* gsrc = Gm + (size_t)hb * (WW * HH);   // 128 contiguous rows of 128 floats
  const float* rsrc = Rm + ((size_t)b * CC + c0) * HH;
  const float* csrc = Cm + ((size_t)b * CC + c0) * WW;
  {
    const float* g0 = gsrc + (size_t)(wv * 16) * 128 + lane * 4;
    float* l0 = Gs + (wv * 16) * PITCH + lane * 4;
#pragma unroll
    for (int r = 0; r < 16; ++r) {        // wave wv: G rows wv*16 .. wv*16+15
      asyncRow128(g0, l0);
      g0 += 128; l0 += PITCH;
    }
  }
  {
    const float* r0 = rsrc + (wv * 2) * 128 + lane * 4;
    const float* c0s = csrc + (wv * 2) * 128 + lane * 4;
    float* lr = Ar + (wv * 2) * PITCH + lane * 4;
    float* lc = Ac + (wv * 2) * PITCH + lane * 4;
#pragma unroll
    for (int r = 0; r < 2; ++r) {         // wave wv: A rows 2wv, 2wv+1
      asyncRow128(r0, lr);
      asyncRow128(c0s, lc);
      r0 += 128; c0s += 128; lr += PITCH; lc += PITCH;
    }
  }
  if (tid < 16) {
    int cc = b * CC + c0 + tid;
    achsig[tid] = a_chan[cc] * sig_ch[cc];
    tbv[tid] = (sm_c[cc] + sig_h[b * HH + hb]) * topo[(size_t)cc * HH + hb];
  }
  asyncWaitAll();
  __syncthreads();

  const int l15 = lane & 15;
  const int koff = (lane >> 4) << 1;     // K sub-offset per WMMA f32 16x16x4 layout
  const int wcol = wv * 16 + l15;        // pixel column within the h row
  const float arow_s = a_row[b * HH + hb];
  const float acolv = a_col[b * WW + wcol];
  const float resv = res_scale[0];

  v8f crow = {0.f, 0.f, 0.f, 0.f, 0.f, 0.f, 0.f, 0.f};
  v8f ccol = {0.f, 0.f, 0.f, 0.f, 0.f, 0.f, 0.f, 0.f};
  const float* arp2 = Ar + l15 * PITCH + koff;
  const float* acp2 = Ac + l15 * PITCH + koff;
  const float* gp2  = Gs + wcol * PITCH + koff;
#pragma unroll
  for (int k = 0; k < 128; k += 4) {
    v2f a_r, a_c, bg;
    a_r[0] = arp2[k]; a_r[1] = arp2[k + 1];
    a_c[0] = acp2[k]; a_c[1] = acp2[k + 1];
    bg[0]  = gp2[k];  bg[1]  = gp2[k + 1];
    crow = __builtin_amdgcn_wmma_f32_16x16x4_f32(false, a_r, false, bg, (short)0, crow, false, false);
    ccol = __builtin_amdgcn_wmma_f32_16x16x4_f32(false, a_c, false, bg, (short)0, ccol, false, false);
  }

  const int hi = lane >> 4;              // D: lanes 16-31 hold M = j+8
#pragma unroll
  for (int j = 0; j < 8; ++j) {
    const int ci = j + hi * 8;
    const size_t off = ((size_t)(b * CC + c0 + ci)) * (HH * WW) + (size_t)hb * WW + wcol;
    const float xv = x[off];
    out[off] = resv * xv + arow_s * crow[j] + acolv * ccol[j]
             + achsig[ci] * xv + arow_s * acolv * tbv[ci];
  }
}

extern "C" void kernel_launch(void* const* d_in, const int* in_sizes, int n_in,
                              void* d_out, int out_size, void* d_ws, size_t ws_size,
                              hipStream_t stream) {
  (void)in_sizes; (void)n_in; (void)out_size; (void)ws_size;
  const float* x         = (const float*)d_in[0];
  const float* U         = (const float*)d_in[1];
  const float* S         = (const float*)d_in[2];
  const float* V         = (const float*)d_in[3];
  const float* Gm        = (const float*)d_in[4];
  const float* w_row     = (const float*)d_in[5];
  const float* w_col     = (const float*)d_in[6];
  const float* boast     = (const float*)d_in[7];
  const float* res_scale = (const float*)d_in[8];
  const float* reduce_w  = (const float*)d_in[9];
  const float* gcn_w     = (const float*)d_in[10];
  const float* gcn_b     = (const float*)d_in[11];
  const float* expand_w  = (const float*)d_in[12];

  float* w = (float*)d_ws;
  float* Xnode   = w;                    // B*C*H   = 262144
  float* topo    = Xnode + 262144;       // B*C*H   = 262144
  float* att_row = topo + 262144;        // B*H*W   = 131072
  float* att_col = att_row + 131072;     // B*H*W   = 131072
  float* Rm      = att_col + 131072;     // B*C*H   = 262144
  float* Cm      = Rm + 262144;          // B*C*W   = 262144
  float* a_row   = Cm + 262144;          // 1024
  float* a_col   = a_row + 1024;         // 1024
  float* a_chan  = a_col + 1024;         // 2048
  float* sig_ch  = a_chan + 2048;        // 2048
  float* sm_c    = sig_ch + 2048;        // 2048
  float* sig_h   = sm_c + 2048;          // 1024  (total ~5.03 MiB)

  k1_rowmean<<<BB * CC, 256, 0, stream>>>(x, Xnode);

  size_t sm2 = 23760 * sizeof(float);    // ~92.8 KB dynamic LDS
  k2_small<<<BB, 128, sm2, stream>>>(U, S, V, w_row, w_col, boast, reduce_w, gcn_w, gcn_b,
                                     expand_w, Xnode, att_row, att_col, topo,
                                     a_row, a_col, a_chan, sig_ch, sm_c, sig_h);

  k3_attred<<<BB * CC, 256, 0, stream>>>(x, att_row, att_col, Rm, Cm);

  size_t sm4sz = (size_t)(128 * 132 + 2 * 16 * 132 + 32) * sizeof(float); // ~82.6 KB
  k4_gemm_out<<<dim3(128, 16, 8), 256, sm4sz, stream>>>(x, Gm, Rm, Cm, topo, a_row, a_col,
                                                        a_chan, sig_ch, sm_c, sig_h, res_scale,
                                                        (float*)d_out);
}